// CausalityMatrix_45938970198714
// MI455X (gfx1250) — compile-verified
//
#include <hip/hip_runtime.h>

typedef __attribute__((ext_vector_type(2))) float v2f;
typedef __attribute__((ext_vector_type(8))) float v8f;

#define BATCH 4
#define MCH   64
#define KSP   196   // 14*14
#define EPSV  1e-9f

// One block per batch. 256 threads = 8 wave32.
// Phase 1: each wave reduces 8 rows (K=196) -> rn[m], w[m] in LDS.
// Phase 2: 16 output tiles (4x4 grid of 16x16) spread 2-per-wave, each tile
//          computed as a rank-1 v_wmma_f32_16x16x4_f32 (u in K=0 col, v in K=0 row).
__global__ __launch_bounds__(256)
void causality_fused_kernel(const float* __restrict__ x,
                            const float* __restrict__ p_num_ptr,
                            const float* __restrict__ p_den_ptr,
                            float* __restrict__ out) {
    __shared__ float s_rn[MCH];  // Lehmer ratio per channel (numerator factor)
    __shared__ float s_w[MCH];   // rn / den per channel (second factor)

    const int b    = blockIdx.x;
    const int lane = threadIdx.x & 31;
    const int wave = threadIdx.x >> 5;

    const float pn = p_num_ptr[0];
    const float pd = p_den_ptr[0];

    // ---------------- Phase 1: per-row Lehmer reductions ----------------
    for (int rr = 0; rr < 8; ++rr) {
        const int m = wave * 8 + rr;                     // 0..63
        const float* xr = x + ((size_t)b * MCH + m) * KSP;

        float s_a = 0.0f, s_a1 = 0.0f, s_d = 0.0f, s_d1 = 0.0f;
        for (int k = lane; k < KSP; k += 32) {
            float xv = xr[k];
            float xe = (xv == 0.0f) ? EPSV : xv;         // strictly positive
            float t  = __builtin_amdgcn_logf(xe);        // v_log_f32: log2(x)
            float ap = __builtin_amdgcn_exp2f(pn * t);   // x^pn
            float dp = __builtin_amdgcn_exp2f(pd * t);   // x^pd
            float ri = __builtin_amdgcn_rcpf(xe);        // 1/x
            s_a  += ap;
            s_a1 += ap * ri;                             // x^(pn-1)
            s_d  += dp;
            s_d1 += dp * ri;                             // x^(pd-1)
        }
        // wave32 butterfly reduction
        #pragma unroll
        for (int msk = 16; msk >= 1; msk >>= 1) {
            s_a  += __shfl_xor(s_a,  msk, 32);
            s_a1 += __shfl_xor(s_a1, msk, 32);
            s_d  += __shfl_xor(s_d,  msk, 32);
            s_d1 += __shfl_xor(s_d1, msk, 32);
        }
        if (lane == 0) {
            float r = s_a / s_a1;
            if (r != r) r = 0.0f;                        // nan_to_zero(numerator factor)
            float den = s_d / s_d1;
            if (den != den) den = 0.0f;                  // nan_to_zero(denominator)
            float ww = r / den;
            if (ww != ww) ww = 0.0f;
            s_rn[m] = r;
            s_w[m]  = ww;
        }
    }
    __syncthreads();

    // ---------------- Phase 2: rank-1 outer product via WMMA ----------------
    const bool lo  = (lane < 16);
    const int  l15 = lane & 15;
    float* o = out + (size_t)b * MCH * MCH;

    #pragma unroll
    for (int t = 0; t < 2; ++t) {
        const int p  = wave * 2 + t;      // 0..15 tile id
        const int mi = p >> 2;            // row-tile
        const int ni = p & 3;             // col-tile

        // A (16x4, f32): lanes 0-15 carry K=0 (VGPR0) / K=1 (VGPR1); lanes 16-31 K=2/K=3.
        // Only K=0 column is nonzero -> D = u ⊗ v.
        v2f a;
        a.x = lo ? s_rn[mi * 16 + l15] : 0.0f;
        a.y = 0.0f;
        // B (4x16, f32): lanes 0-15 = N=0..15 with K=0 in VGPR0; rest zero.
        v2f bb;
        bb.x = lo ? s_w[ni * 16 + l15] : 0.0f;
        bb.y = 0.0f;

        v8f c = {};
        // 8 args: (neg_a, A, neg_b, B, c_mod, C, reuse_a, reuse_b)
        c = __builtin_amdgcn_wmma_f32_16x16x4_f32(
                false, a, false, bb, (short)0, c, false, false);

        // D layout: VGPR r -> M=r (lanes 0-15) / M=r+8 (lanes 16-31), N = lane&15.
        const int radd = lo ? 0 : 8;
        #pragma unroll
        for (int r = 0; r < 8; ++r) {
            float d = c[r];
            if (d != d) d = 0.0f;                        // final nan_to_zero
            o[(size_t)(mi * 16 + r + radd) * MCH + (ni * 16 + l15)] = d;
        }
    }
}

extern "C" void kernel_launch(void* const* d_in, const int* in_sizes, int n_in,
                              void* d_out, int out_size, void* d_ws, size_t ws_size,
                              hipStream_t stream) {
    (void)in_sizes; (void)n_in; (void)out_size; (void)d_ws; (void)ws_size;
    const float* x  = (const float*)d_in[0];
    const float* pn = (const float*)d_in[1];
    const float* pd = (const float*)d_in[2];
    float* out = (float*)d_out;

    causality_fused_kernel<<<BATCH, 256, 0, stream>>>(x, pn, pd, out);
}